// AdaptiveSelection_34488587387150
// MI455X (gfx1250) — compile-verified
//
#include <hip/hip_runtime.h>
#include <hip/hip_bf16.h>

typedef float v2f __attribute__((ext_vector_type(2)));
typedef float v8f __attribute__((ext_vector_type(8)));

#define DIM     1024
#define NCLUST  8
#define NPER    4096
#define QDIM    128
#define TOPK    128
#define NSAMPLE 1024

// ws layout (float units):
//  [0,     8192)  wvec   : per-cluster score vector w_c = q_w^T (q_w key_c + q_b)   (8 x 1024)
//  [8192,  8200)  bias   : per-cluster scalar q_b . qk_c                            (8)
//  [8200, 40968)  logitsA: logits then softmax A                                    (8 x 4096)
//  [40968,49160)  svec   : S = A^T feats                                            (8 x 1024)
//  [49160,50184)  idxbuf : top-k indices (as int)                                   (8 x 128)

// ---------------------------------------------------------------- prep: qk, wvec, bias, zero S
__global__ void prep_kernel(const float* __restrict__ q_w, const float* __restrict__ q_b,
                            const float* __restrict__ key_feats, float* __restrict__ ws) {
    __shared__ float qk[QDIM];
    const int c = blockIdx.x;
    const int t = threadIdx.x;            // 128 threads
    {   // qk[q] = q_w[q,:].key_c + q_b[q]
        const float* qr = q_w + (size_t)t * DIM;
        const float* kf = key_feats + (size_t)c * DIM;
        float acc = 0.f;
        for (int d = 0; d < DIM; ++d) acc += qr[d] * kf[d];
        qk[t] = acc + q_b[t];
    }
    __syncthreads();
    float* wvec = ws + (size_t)c * DIM;
    for (int d = t; d < DIM; d += 128) {
        float acc = 0.f;
        for (int q = 0; q < QDIM; ++q) acc += q_w[(size_t)q * DIM + d] * qk[q];
        wvec[d] = acc;
    }
    float* svec = ws + 40968 + (size_t)c * DIM;       // zero S for the atomic pass
    for (int d = t; d < DIM; d += 128) svec[d] = 0.f;
    if (t == 0) {
        float acc = 0.f;
        for (int q = 0; q < QDIM; ++q) acc += q_b[q] * qk[q];
        ws[8192 + c] = acc;
    }
}

// ---------------------------------------------------------------- logits[c,r] = (feats[r].w_c + b_c)/sqrt(128)
// HBM-bound pass #1: one wave per row, float4 (b128) loads.
__global__ void logits_kernel(const float* __restrict__ feats, const float* __restrict__ ws,
                              float* __restrict__ logits) {
    __shared__ float wv[DIM];
    const int c = blockIdx.y;
    const int chunk = blockIdx.x;         // 16 chunks of 256 rows
    const int t = threadIdx.x;            // 256 threads = 8 waves
    for (int d = t; d < DIM; d += 256) wv[d] = ws[(size_t)c * DIM + d];
    __syncthreads();
    const float bias  = ws[8192 + c];
    const float scale = 0.08838834764831845f;   // 1/sqrt(QDIM)
    const int wave = t >> 5, lane = t & 31;
    const float4* wv4 = (const float4*)wv;
    const float* fc = feats + (size_t)c * NPER * DIM;
    const int r0 = chunk * 256;
    for (int r = r0 + wave; r < r0 + 256; r += 8) {       // one wave per row
        const float4* row4 = (const float4*)(fc + (size_t)r * DIM);
        float acc = 0.f;
        #pragma unroll
        for (int jj = 0; jj < 8; ++jj) {                  // 8 * 32 lanes * float4 = 1024
            const float4 f = row4[jj * 32 + lane];
            const float4 w = wv4[jj * 32 + lane];
            acc += f.x * w.x + f.y * w.y + f.z * w.z + f.w * w.w;
        }
        for (int m = 16; m >= 1; m >>= 1) acc += __shfl_xor(acc, m, 32);
        if (lane == 0) logits[(size_t)c * NPER + r] = (acc + bias) * scale;
    }
}

// ---------------------------------------------------------------- softmax over 4096 + iterative top-128
__global__ void softmax_topk_kernel(float* __restrict__ logitsA, int* __restrict__ idxbuf) {
    __shared__ float vals[NPER];
    __shared__ float redv[256];
    __shared__ int   redi[256];
    const int c = blockIdx.x, t = threadIdx.x;            // 256 threads
    float* gl = logitsA + (size_t)c * NPER;
    float lmax = -3.4e38f;
    for (int i = t; i < NPER; i += 256) { float v = gl[i]; vals[i] = v; lmax = fmaxf(lmax, v); }
    redv[t] = lmax; __syncthreads();
    for (int s = 128; s >= 1; s >>= 1) { if (t < s) redv[t] = fmaxf(redv[t], redv[t + s]); __syncthreads(); }
    const float mx = redv[0]; __syncthreads();
    float lsum = 0.f;
    for (int i = t; i < NPER; i += 256) { float e = __expf(vals[i] - mx); vals[i] = e; lsum += e; }
    redv[t] = lsum; __syncthreads();
    for (int s = 128; s >= 1; s >>= 1) { if (t < s) redv[t] += redv[t + s]; __syncthreads(); }
    const float inv = 1.f / redv[0]; __syncthreads();
    for (int i = t; i < NPER; i += 256) { float a = vals[i] * inv; vals[i] = a; gl[i] = a; }
    __syncthreads();
    // iterative argmax, tie-break lower index (matches lax.top_k ordering)
    for (int it = 0; it < TOPK; ++it) {
        float bv = -1.f; int bi = 0;
        const int base = t * 16;
        for (int j = 0; j < 16; ++j) {
            float v = vals[base + j];
            if (v > bv) { bv = v; bi = base + j; }
        }
        redv[t] = bv; redi[t] = bi; __syncthreads();
        for (int s = 128; s >= 1; s >>= 1) {
            if (t < s) {
                float v2 = redv[t + s]; int i2 = redi[t + s];
                if (v2 > redv[t] || (v2 == redv[t] && i2 < redi[t])) { redv[t] = v2; redi[t] = i2; }
            }
            __syncthreads();
        }
        if (t == 0) { idxbuf[c * TOPK + it] = redi[0]; vals[redi[0]] = -2.f; }
        __syncthreads();
    }
}

// ---------------------------------------------------------------- gather top rows -> selected output
__global__ void gather_kernel(const float* __restrict__ feats, const int* __restrict__ idxbuf,
                              float* __restrict__ out) {
    const int r = blockIdx.x;             // 0..1023
    const int c = r >> 7, j = r & 127;
    const int src = idxbuf[c * TOPK + j];
    const float4* s4 = (const float4*)(feats + ((size_t)c * NPER + src) * DIM);
    float4* d4 = (float4*)(out + (size_t)r * DIM);
    d4[threadIdx.x] = s4[threadIdx.x];    // 256 threads x float4 = 1024 floats
}

// ---------------------------------------------------------------- S[c,:] += sum_r A[c,r]*feats[c,r,:]
// Pass #2 over feats (L2-resident: 128MB < 192MB L2). Each thread owns a 4-col float4 strip.
__global__ void weighted_kernel(const float* __restrict__ feats, const float* __restrict__ A,
                                float* __restrict__ svec) {
    const int c = blockIdx.y, chunk = blockIdx.x, t = threadIdx.x;   // 256 threads, 16 chunks
    const float4* fc4 = (const float4*)(feats + (size_t)c * NPER * DIM);
    const float* ac = A + (size_t)c * NPER;
    float4 acc = {0.f, 0.f, 0.f, 0.f};
    const int r0 = chunk * 256;
    for (int r = r0; r < r0 + 256; ++r) {
        const float a = ac[r];
        const float4 f = fc4[(size_t)r * 256 + t];    // coalesced b128
        acc.x += a * f.x; acc.y += a * f.y; acc.z += a * f.z; acc.w += a * f.w;
    }
    float* sp = svec + (size_t)c * DIM + 4 * t;
    atomicAdd(sp + 0, acc.x);
    atomicAdd(sp + 1, acc.y);
    atomicAdd(sp + 2, acc.z);
    atomicAdd(sp + 3, acc.w);
}

// ---------------------------------------------------------------- fusion = S @ v_w^T + v_b via fp32 WMMA
// M=16 (8 clusters + 8 zero pad), N=1024, K=1024. One wave per 16-wide N tile.
// A 16x4 f32 layout: lanes 0-15 -> M=lane,K={k,k+1}; lanes 16-31 -> M=lane-16,K={k+2,k+3}.
// B 4x16 f32 layout (mirror): lanes 0-15 -> N=lane,K={k,k+1}; lanes 16-31 -> N=lane-16,K={k+2,k+3}.
// Pad rows handled branch-free with a 0/1 lane mask so EXEC stays all-1s through the loop.
__global__ void fusion_wmma_kernel(const float* __restrict__ svec, const float* __restrict__ v_w,
                                   const float* __restrict__ v_b, float* __restrict__ out_fus) {
    const int lane  = threadIdx.x & 31;
    const int nbase = blockIdx.x * 16;
    const int half  = lane >> 4;          // 0: K lo pair, 1: K hi pair
    const int ml    = lane & 15;          // row (A) / col (B) within half
    const float amask = (ml < 8) ? 1.0f : 0.0f;               // zero pad rows 8..15
    const float* arow = svec + (size_t)(ml & 7) * DIM;        // safe clamped row (S is L2-resident)
    const float* brow = v_w + (size_t)(nbase + ml) * DIM;     // B[k][n] = v_w[n][k]
    v8f acc = {0.f, 0.f, 0.f, 0.f, 0.f, 0.f, 0.f, 0.f};
    for (int k = 0; k < DIM; k += 4) {
        const int kk = k + half * 2;
        v2f a = *(const v2f*)(arow + kk);
        a.x *= amask; a.y *= amask;
        const v2f b = *(const v2f*)(brow + kk);
        acc = __builtin_amdgcn_wmma_f32_16x16x4_f32(
                  /*neg_a=*/false, a, /*neg_b=*/false, b,
                  /*c_mod=*/(short)0, acc, /*reuse_a=*/false, /*reuse_b=*/false);
    }
    if (lane < 16) {
        const float vb = v_b[nbase + lane];
        #pragma unroll
        for (int cc = 0; cc < NCLUST; ++cc)
            out_fus[(size_t)cc * DIM + nbase + lane] = acc[cc] + vb;
    }
}

// ---------------------------------------------------------------- launch
extern "C" void kernel_launch(void* const* d_in, const int* in_sizes, int n_in,
                              void* d_out, int out_size, void* d_ws, size_t ws_size,
                              hipStream_t stream) {
    const float* feats     = (const float*)d_in[0];   // 8 x 4096 x 1024
    const float* key_feats = (const float*)d_in[1];   // 8 x 1 x 1024
    const float* q_w       = (const float*)d_in[2];   // 128 x 1024
    const float* q_b       = (const float*)d_in[3];   // 128
    const float* v_w       = (const float*)d_in[4];   // 1024 x 1024
    const float* v_b       = (const float*)d_in[5];   // 1024

    float* ws      = (float*)d_ws;
    float* logitsA = ws + 8200;
    float* svec    = ws + 40968;
    int*   idxbuf  = (int*)(ws + 49160);
    float* out     = (float*)d_out;                   // [0,1048576): selected; [1048576,+8192): fusion

    prep_kernel<<<NCLUST, 128, 0, stream>>>(q_w, q_b, key_feats, ws);
    logits_kernel<<<dim3(16, NCLUST), 256, 0, stream>>>(feats, ws, logitsA);
    softmax_topk_kernel<<<NCLUST, 256, 0, stream>>>(logitsA, idxbuf);
    gather_kernel<<<NSAMPLE, 256, 0, stream>>>(feats, idxbuf, out);
    weighted_kernel<<<dim3(16, NCLUST), 256, 0, stream>>>(feats, logitsA, svec);
    fusion_wmma_kernel<<<DIM / 16, 32, 0, stream>>>(svec, v_w, v_b, out + (size_t)NSAMPLE * DIM);
}